// ImageEncoder2_4681514353166
// MI455X (gfx1250) — compile-verified
//
#include <hip/hip_runtime.h>

typedef float v2f __attribute__((ext_vector_type(2)));
typedef float v4f __attribute__((ext_vector_type(4)));
typedef float v8f __attribute__((ext_vector_type(8)));

#define B_    16
#define N_    36
#define D_    512
#define NT_   30
#define NSPAN 630   // N*(N-1)/2 = 36*35/2

// -------------------------------------------------------------------------
// Kernel 1: X[576,512] = F[576,512] @ W[512,512] + bias   (f32 WMMA 16x16x4)
// One wave32 per 16x16 output tile. 36 M-tiles x 32 N-tiles = 1152 waves.
// -------------------------------------------------------------------------
__global__ __launch_bounds__(256) void ie2_gemm_wmma(
    const float* __restrict__ F, const float* __restrict__ W,
    const float* __restrict__ bias, float* __restrict__ X) {
  const int tid  = threadIdx.x;
  const int lane = tid & 31;
  const int wave = blockIdx.x * 8 + (tid >> 5);   // 0..1151
  const int tile_m = wave >> 5;                   // 0..35
  const int tile_n = wave & 31;                   // 0..31

  const int mrow  = tile_m * 16 + (lane & 15);    // A-frag row (M = lane%16)
  const int khalf = (lane >> 4) * 2;              // K sub-offset: 0 or 2
  const int ncol  = tile_n * 16 + (lane & 15);    // B/D column (N = lane%16)

  const float* frow = F + (size_t)mrow * D_;

  v8f acc = {};
  #pragma unroll 4
  for (int k0 = 0; k0 < D_; k0 += 4) {
    // A 16x4: lane holds (M=lane%16, K=khalf+j) -> contiguous float2 (b64 load)
    v2f a = *(const v2f*)(frow + k0 + khalf);
    // B 4x16 (W row-major [K][N]): lane holds (K=khalf+j, N=ncol)
    v2f b;
    b.x = W[(size_t)(k0 + khalf + 0) * D_ + ncol];
    b.y = W[(size_t)(k0 + khalf + 1) * D_ + ncol];
    acc = __builtin_amdgcn_wmma_f32_16x16x4_f32(
        /*neg_a=*/false, a, /*neg_b=*/false, b,
        /*c_mod=*/(short)0, acc, /*reuse_a=*/false, /*reuse_b=*/false);
  }

  // D layout: lane l holds D[r + 8*(l/16)][l%16] in acc[r]
  const float bv    = bias[ncol];
  const int   mbase = tile_m * 16 + 8 * (lane >> 4);
  #pragma unroll
  for (int r = 0; r < 8; ++r)
    X[(size_t)(mbase + r) * D_ + ncol] = acc[r] + bv;
}

// -------------------------------------------------------------------------
// Kernel 2: CS[b][0]=0; CS[b][i+1][d] = sum_{r<=i} X[b][r][d]
// One thread per (b,d): 16*512 = 8192 threads, fully coalesced over d.
// -------------------------------------------------------------------------
__global__ __launch_bounds__(256) void ie2_cumsum(
    const float* __restrict__ X, float* __restrict__ CS) {
  const int gid = blockIdx.x * 256 + threadIdx.x;  // 0..8191
  const int b = gid >> 9;
  const int d = gid & 511;
  float run = 0.f;
  CS[((size_t)b * (N_ + 1)) * D_ + d] = 0.f;
  #pragma unroll 4
  for (int i = 0; i < N_; ++i) {
    run += X[((size_t)b * N_ + i) * D_ + d];
    CS[((size_t)b * (N_ + 1) + i + 1) * D_ + d] = run;
  }
}

// -------------------------------------------------------------------------
// Kernel 3: per (b, span): diff of two CS rows (L2-resident), l2-normalize,
// stream 30 identical copies with 128-bit nontemporal stores.
// Output [16, 630, 30, 512] f32 = 619 MB -> the HBM-bound stage (~27us floor).
// -------------------------------------------------------------------------
__global__ __launch_bounds__(256) void ie2_span(
    const float* __restrict__ CS, float* __restrict__ OUT) {
  const int sidx = blockIdx.x;   // 0..629
  const int b    = blockIdx.y;   // 0..15
  const int tid  = threadIdx.x;

  // Map span index -> (span length k+1, start). Reference order: k=1..N-1,
  // each block has N-k spans starting at 0..N-k-1. Uniform scalar loop.
  int s = sidx, k = 1;
  while (s >= N_ - k) { s -= N_ - k; ++k; }
  const int start = s;

  const float* lo = CS + ((size_t)b * (N_ + 1) + start) * D_;
  const float* hi = CS + ((size_t)b * (N_ + 1) + start + k + 1) * D_;

  __shared__ float row[D_];
  __shared__ float red[256];

  const int d0 = tid * 2;
  v2f v;
  v.x = hi[d0]     - lo[d0];
  v.y = hi[d0 + 1] - lo[d0 + 1];
  red[tid] = v.x * v.x + v.y * v.y;
  __syncthreads();
  #pragma unroll
  for (int off = 128; off > 0; off >>= 1) {
    if (tid < off) red[tid] += red[tid + off];
    __syncthreads();
  }
  const float inv = rsqrtf(red[0]);
  row[d0]     = v.x * inv;
  row[d0 + 1] = v.y * inv;
  __syncthreads();

  // 30 copies * 128 float4 = 3840 b128 stores; 15 per thread, NT hint.
  float* outBase = OUT + ((size_t)(b * NSPAN + sidx)) * (NT_ * D_);
  #pragma unroll
  for (int i = 0; i < 15; ++i) {
    const int idx = tid + i * 256;     // 0..3839
    const int tt  = idx >> 7;          // copy index 0..29
    const int d4  = idx & 127;         // float4 index within row
    v4f val = *(const v4f*)(&row[d4 * 4]);
    __builtin_nontemporal_store(val, (v4f*)(outBase + (size_t)tt * D_ + d4 * 4));
  }
}

// -------------------------------------------------------------------------
extern "C" void kernel_launch(void* const* d_in, const int* in_sizes, int n_in,
                              void* d_out, int out_size, void* d_ws, size_t ws_size,
                              hipStream_t stream) {
  const float* features = (const float*)d_in[0];
  // d_in[1] = img_lengths (int64, all == N) -- structurally unused
  const float* W1 = (const float*)d_in[2];
  const float* b1 = (const float*)d_in[3];
  float* out = (float*)d_out;

  float* X  = (float*)d_ws;                               // 576*512 f32 = 1.18 MB
  float* CS = X + (size_t)B_ * N_ * D_;                   // 16*37*512 f32 = 1.21 MB

  // 1) GEMM via f32 WMMA: 1152 tiles, 8 waves/block -> 144 blocks
  ie2_gemm_wmma<<<144, 256, 0, stream>>>(features, W1, b1, X);
  // 2) prefix sums: 8192 threads
  ie2_cumsum<<<32, 256, 0, stream>>>(X, CS);
  // 3) span diff + normalize + NT broadcast store: one block per (b, span)
  ie2_span<<<dim3(NSPAN, B_), 256, 0, stream>>>(CS, out);
}